// MarginRatioDistribution_32676111188447
// MI455X (gfx1250) — compile-verified
//
#include <hip/hip_runtime.h>
#include <math.h>

#define BN 4096
#define CN 1000
#define DN 2048
#define DIST_STRIDE 1024
#define DATA_SCALING (1.0f / 0.224f)
#define LDS_STRIDE 36   // gcd-free vs 64 banks for 16-lane row-strided reads; rows 16B aligned

typedef float v2f __attribute__((ext_vector_type(2)));
typedef float v8f __attribute__((ext_vector_type(8)));

// ---------------- Kernel 1: per-row max + argmax of prediction (BN x CN) -----
__global__ __launch_bounds__(256)
void rowArgmaxKernel(const float* __restrict__ pred,
                     int* __restrict__ j0, float* __restrict__ ymax) {
  __shared__ float sval[256];
  __shared__ int   sidx[256];
  const int row = blockIdx.x;
  const int t = threadIdx.x;
  const float* p = pred + (size_t)row * CN;
  float best = -__builtin_inff();
  int bidx = 0;
  for (int c = t; c < CN; c += 256) {
    float v = p[c];
    if (v > best) { best = v; bidx = c; }
  }
  sval[t] = best; sidx[t] = bidx;
  __syncthreads();
  for (int s = 128; s > 0; s >>= 1) {
    if (t < s) {
      float ov = sval[t + s]; int oi = sidx[t + s];
      if (ov > sval[t] || (ov == sval[t] && oi < sidx[t])) { sval[t] = ov; sidx[t] = oi; }
    }
    __syncthreads();
  }
  if (t == 0) { j0[row] = sidx[0]; ymax[row] = sval[0]; }
}

// ---------------- Kernel 2: row squared norms of W (CN x DN) -----------------
__global__ __launch_bounds__(256)
void rowSqNormKernel(const float* __restrict__ W, float* __restrict__ sqw) {
  __shared__ float s[256];
  const int row = blockIdx.x;
  const int t = threadIdx.x;
  const float* p = W + (size_t)row * DN;
  float acc = 0.0f;
  for (int c = t; c < DN; c += 256) { float v = p[c]; acc += v * v; }
  s[t] = acc;
  __syncthreads();
  for (int st = 128; st > 0; st >>= 1) { if (t < st) s[t] += s[t + st]; __syncthreads(); }
  if (t == 0) sqw[row] = s[0];
}

// ---------------- Kernel 3: Gram matrix via f32 WMMA, fused distance ---------
// Each workgroup: 64x64 block of G = W @ W^T, K=DN staged in 64x32 LDS chunks.
// 8 waves; wave w owns tile row (w&3) and tile cols {2*(w>>2), 2*(w>>2)+1}.
__global__ __launch_bounds__(256)
void gramDistKernel(const float* __restrict__ W, const float* __restrict__ sqw,
                    const float* __restrict__ Kp, float* __restrict__ dist) {
  __shared__ float lA[64 * LDS_STRIDE];
  __shared__ float lB[64 * LDS_STRIDE];

  const int rowBase = blockIdx.y * 64;
  const int colBase = blockIdx.x * 64;
  const int t    = threadIdx.x;
  const int lane = t & 31;
  const int wid  = t >> 5;
  const int tr   = wid & 3;          // tile row within 4x4 tile grid
  const int tc0  = (wid >> 2) * 2;   // first of two tile cols

  v8f acc0 = {};
  v8f acc1 = {};

  // cooperative-load mapping: thread t -> row t>>2 (0..63), cols (t&3)*8 .. +7
  const int lr = t >> 2;
  const int lc = (t & 3) * 8;
  const float kscale = Kp[0] * DATA_SCALING;

  const int mrow = lane & 15;          // M (A) or N (B) within tile
  const int koff = (lane >> 4) << 1;   // lanes 16-31 hold K+2,K+3

  for (int k0 = 0; k0 < DN; k0 += 32) {
    // ---- stage 64x32 chunks of A-rows and B-rows into LDS (zero-pad OOB) ----
    {
      float4 a0 = {0.f,0.f,0.f,0.f}, a1 = {0.f,0.f,0.f,0.f};
      const int ga = rowBase + lr;
      if (ga < CN) {
        const float* src = W + (size_t)ga * DN + k0 + lc;
        a0 = *(const float4*)src;
        a1 = *(const float4*)(src + 4);
      }
      *(float4*)&lA[lr * LDS_STRIDE + lc]     = a0;
      *(float4*)&lA[lr * LDS_STRIDE + lc + 4] = a1;

      float4 b0 = {0.f,0.f,0.f,0.f}, b1 = {0.f,0.f,0.f,0.f};
      const int gb = colBase + lr;
      if (gb < CN) {
        const float* src = W + (size_t)gb * DN + k0 + lc;
        b0 = *(const float4*)src;
        b1 = *(const float4*)(src + 4);
      }
      *(float4*)&lB[lr * LDS_STRIDE + lc]     = b0;
      *(float4*)&lB[lr * LDS_STRIDE + lc + 4] = b1;
    }
    __syncthreads();

    // ---- 8 WMMA steps per chunk per tile (K advances by 4) ----
    #pragma unroll
    for (int kk = 0; kk < 32; kk += 4) {
      v2f a  = *(const v2f*)&lA[(tr  * 16 + mrow) * LDS_STRIDE + kk + koff];
      v2f b0 = *(const v2f*)&lB[(tc0 * 16 + mrow) * LDS_STRIDE + kk + koff];
      v2f b1 = *(const v2f*)&lB[((tc0 + 1) * 16 + mrow) * LDS_STRIDE + kk + koff];
      acc0 = __builtin_amdgcn_wmma_f32_16x16x4_f32(
          false, a, false, b0, (short)0, acc0, false, false);
      acc1 = __builtin_amdgcn_wmma_f32_16x16x4_f32(
          false, a, false, b1, (short)0, acc1, false, false);
    }
    __syncthreads();
  }

  // ---- write fused distances: dist[j][c] = k*sqrt(max(|Wj|^2+|Wc|^2-2G,0)) --
  const int n   = lane & 15;
  const int mhi = (lane >> 4) * 8;   // C/D layout: VGPR r -> M = r (+8 for hi lanes)
  #pragma unroll
  for (int r = 0; r < 8; ++r) {
    const int j  = rowBase + tr * 16 + r + mhi;
    const int c0 = colBase + tc0 * 16 + n;
    const int c1 = colBase + (tc0 + 1) * 16 + n;
    if (j < CN) {
      const float sj = sqw[j];
      if (c0 < CN) {
        float v = sj + sqw[c0] - 2.0f * acc0[r];
        dist[(size_t)j * DIST_STRIDE + c0] = kscale * sqrtf(fmaxf(v, 0.0f));
      }
      if (c1 < CN) {
        float v = sj + sqw[c1] - 2.0f * acc1[r];
        dist[(size_t)j * DIST_STRIDE + c1] = kscale * sqrtf(fmaxf(v, 0.0f));
      }
    }
  }
}

// ---------------- Kernel 4: streaming ratio-min per row ----------------------
__global__ __launch_bounds__(256)
void ratioMinKernel(const float* __restrict__ pred,
                    const int* __restrict__ j0,
                    const float* __restrict__ ymax,
                    const float* __restrict__ dist,
                    float* __restrict__ out) {
  __shared__ float s[256];
  const int row = blockIdx.x;
  const int t = threadIdx.x;
  const int jj = j0[row];
  const float ym = ymax[row];
  const float* p  = pred + (size_t)row * CN;
  const float* dr = dist + (size_t)jj * DIST_STRIDE;
  float best = __builtin_inff();
  for (int c = t; c < CN; c += 256) {
    if (c == jj) continue;                     // onehot column -> +inf in ref
    float ratio = (ym - p[c]) / dr[c];
    best = fminf(best, ratio);
  }
  s[t] = best;
  __syncthreads();
  for (int st = 128; st > 0; st >>= 1) {
    if (t < st) s[t] = fminf(s[t], s[t + st]);
    __syncthreads();
  }
  if (t == 0) out[row] = s[0];
}

// ---------------- launcher ---------------------------------------------------
extern "C" void kernel_launch(void* const* d_in, const int* in_sizes, int n_in,
                              void* d_out, int out_size, void* d_ws, size_t ws_size,
                              hipStream_t stream) {
  const float* pred = (const float*)d_in[0];   // (4096, 1000) f32
  const float* W    = (const float*)d_in[1];   // (1000, 2048) f32
  const float* Kp   = (const float*)d_in[2];   // (1,) f32
  float* out = (float*)d_out;                  // (4096,) f32

  // workspace layout (floats): [j0: BN][ymax: BN][sqw: 1024][dist: 1024*1024]
  int*   j0   = (int*)d_ws;
  float* ymax = (float*)d_ws + BN;
  float* sqw  = (float*)d_ws + 2 * BN;
  float* dist = (float*)d_ws + 2 * BN + 1024;

  rowArgmaxKernel<<<BN, 256, 0, stream>>>(pred, j0, ymax);
  rowSqNormKernel<<<CN, 256, 0, stream>>>(W, sqw);
  dim3 grid(16, 16);                            // 16*64 = 1024 covers CN=1000
  gramDistKernel<<<grid, 256, 0, stream>>>(W, sqw, Kp, dist);
  ratioMinKernel<<<BN, 256, 0, stream>>>(pred, j0, ymax, dist, out);
}